// BoxMatchKDD_1941325218101
// MI455X (gfx1250) — compile-verified
//
#include <hip/hip_runtime.h>
#include <hip/hip_bf16.h>
#include <stdint.h>

typedef __attribute__((ext_vector_type(2))) float v2f;
typedef __attribute__((ext_vector_type(8))) float v8f;
typedef unsigned int u32x4 __attribute__((ext_vector_type(4)));
typedef int i32x4 __attribute__((ext_vector_type(4)));
typedef int i32x8 __attribute__((ext_vector_type(8)));

#define N_IMG 64
#define T_BOX 900
#define S_BOX 900
#define C_CLS 91
#define NEGV  (-1e30f)

__device__ __forceinline__ unsigned rfl(unsigned x) {
    return (unsigned)__builtin_amdgcn_readfirstlane((int)x);
}

// Tensor Data Mover: issue one TENSOR_LOAD_TO_LDS per CDNA5 ISA §8 D# layout.
// group0: [1:0]=count=1, [31]=gather_mode, [63:32]=lds_addr,
//         [120:64]=global_addr, [127:126]=type=2
// group1: [17:16]=data_size(2=4B), [63:48]+[79:64]=tensor_dim0,
//         [95:80]+[111:96]=tensor_dim1, [127:112]=tile_dim0,
//         [143:128]=tile_dim1, [207:160]=tensor_dim0_stride (elements)
// groups2/3: zeros (normal) or 16x 16-bit row indices (gather mode)
// This toolchain exposes the 6-arg builtin: extra int32x8 slot is zero-filled.
__device__ __forceinline__ void tdm_issue(
    unsigned lds_addr, unsigned long long ga, int gather,
    unsigned dim0, unsigned dim1, unsigned tile0, unsigned tile1,
    unsigned stride0, i32x4 g2, i32x4 g3)
{
    u32x4 g0;
    g0.x = 1u | (gather ? (1u << 31) : 0u);
    g0.y = lds_addr;
    g0.z = rfl((unsigned)ga);
    g0.w = rfl((unsigned)(ga >> 32)) | (2u << 30);
    i32x8 g1;
    g1[0] = (int)(2u << 16);                                          // 4B elems
    g1[1] = (int)((dim0 & 0xFFFFu) << 16);
    g1[2] = (int)(((dim0 >> 16) & 0xFFFFu) | ((dim1 & 0xFFFFu) << 16));
    g1[3] = (int)(((dim1 >> 16) & 0xFFFFu) | ((tile0 & 0xFFFFu) << 16));
    g1[4] = (int)(tile1 & 0xFFFFu);                                   // tile_dim2=0
    g1[5] = (int)stride0;
    g1[6] = 0;
    g1[7] = 0;
    i32x8 zx = {0, 0, 0, 0, 0, 0, 0, 0};
    __builtin_amdgcn_tensor_load_to_lds(g0, g1, g2, g3, zx, 0);
}

// ---------------- Phase A: IoU + argmax over students ----------------
#define TCH   8
#define CHUNK 113   // ceil(900/8)

__global__ __launch_bounds__(256) void iou_argmax_kernel(
    const float* __restrict__ t_boxes, const float* __restrict__ s_boxes,
    const int* __restrict__ s_valid,
    float* __restrict__ best_iou, int* __restrict__ best_j)
{
    __shared__ __align__(16) float raw[S_BOX * 4];     // TDM-staged [S][4] boxes
    __shared__ float sx0[S_BOX], sy0[S_BOX], sx1[S_BOX], sy1[S_BOX];
    __shared__ float sar[S_BOX], svf[S_BOX];

    const int n     = blockIdx.x / TCH;
    const int chunk = blockIdx.x % TCH;

    if ((threadIdx.x >> 5) == 0) {     // one wave drives the TDM
        unsigned lds = rfl((unsigned)(uintptr_t)&raw[0]);
        unsigned long long ga =
            (unsigned long long)(uintptr_t)(s_boxes + (size_t)n * S_BOX * 4);
        i32x4 z = {0, 0, 0, 0};
        tdm_issue(lds, ga, 0, S_BOX * 4, 1, S_BOX * 4, 1, S_BOX * 4, z, z);
        __builtin_amdgcn_s_wait_tensorcnt(0);
    }
    __syncthreads();

    for (int s = threadIdx.x; s < S_BOX; s += 256) {
        float4 b = reinterpret_cast<const float4*>(raw)[s];
        sx0[s] = b.x; sy0[s] = b.y; sx1[s] = b.z; sy1[s] = b.w;
        sar[s] = (b.z - b.x) * (b.w - b.y);
        svf[s] = s_valid[n * S_BOX + s] ? 1.0f : 0.0f;
    }
    __syncthreads();

    const int wave = threadIdx.x >> 5;
    const int lane = threadIdx.x & 31;

    for (int tl = wave; tl < CHUNK; tl += 8) {
        const int t = chunk * CHUNK + tl;
        if (t >= T_BOX) break;   // wave-uniform
        float4 a = reinterpret_cast<const float4*>(t_boxes)[n * T_BOX + t];
        const float aa = (a.z - a.x) * (a.w - a.y);

        float bv = NEGV; int bj = 0;
        for (int s = lane; s < S_BOX; s += 32) {
            float w = fminf(a.z, sx1[s]) - fmaxf(a.x, sx0[s]);
            float h = fminf(a.w, sy1[s]) - fmaxf(a.y, sy0[s]);
            w = fmaxf(w, 0.0f); h = fmaxf(h, 0.0f);
            float inter = w * h;
            float iou   = inter / (aa + sar[s] - inter);
            float val   = (svf[s] != 0.0f) ? iou : NEGV;
            if (val > bv) { bv = val; bj = s; }
        }
        for (int off = 16; off >= 1; off >>= 1) {
            float ov = __shfl_xor(bv, off, 32);
            int   oj = __shfl_xor(bj, off, 32);
            if (ov > bv || (ov == bv && oj < bj)) { bv = ov; bj = oj; }
        }
        if (lane == 0) { best_iou[n * T_BOX + t] = bv; best_j[n * T_BOX + t] = bj; }
    }
}

// ---------------- Phase B: TDM-staged softmax/KL with f32 WMMA ----------------
#define KD_WAVES 4
#define KD_ROWS  64   // 4 waves * 16 rows
#define KD_BLKS  15   // ceil(900/64)

__global__ __launch_bounds__(128) void kd_wmma_kernel(
    const float* __restrict__ t_logits, const int* __restrict__ t_valid,
    const float* __restrict__ s_logits,
    const float* __restrict__ best_iou, const int* __restrict__ best_j,
    float* __restrict__ per_arr, float* __restrict__ keep_arr)
{
    __shared__ __align__(16) float stage_t[KD_WAVES][16 * C_CLS]; // teacher tile
    __shared__ __align__(16) float stage_s[KD_WAVES][16 * C_CLS]; // gathered students
    __shared__ float lds_d[KD_WAVES][32][8];                      // diag spill

    const int n    = blockIdx.x / KD_BLKS;
    const int blk  = blockIdx.x % KD_BLKS;
    const int wave = threadIdx.x >> 5;
    const int lane = threadIdx.x & 31;
    const int L    = lane & 15;
    const int h    = lane >> 4;

    const int t_base = blk * KD_ROWS + wave * 16;
    const int t_raw  = t_base + L;
    const int t      = (t_raw < T_BOX) ? t_raw : (T_BOX - 1);
    const bool owner = (t_raw < T_BOX) && (h == 0);

    const int j = best_j[n * T_BOX + t];   // lanes 0..15 carry the 16 row indices

    // TDM #1: 16 consecutive teacher rows; rows past tensor_dim1 return zeros.
    {
        int eff_base = t_base, rows = T_BOX - t_base;
        if (rows <= 0) { eff_base = 0; rows = 16; }   // fully-OOB tile: dummy load
        unsigned lds = rfl((unsigned)(uintptr_t)&stage_t[wave][0]);
        unsigned long long ga = (unsigned long long)(uintptr_t)
            (t_logits + ((size_t)n * T_BOX + (size_t)eff_base) * C_CLS);
        i32x4 z = {0, 0, 0, 0};
        tdm_issue(lds, ga, 0, C_CLS, rfl((unsigned)rows), C_CLS, 16, C_CLS, z, z);
    }
    // TDM #2: gather-mode load of the 16 best_j student rows.
    {
        unsigned lds = rfl((unsigned)(uintptr_t)&stage_s[wave][0]);
        unsigned long long ga = (unsigned long long)(uintptr_t)
            (s_logits + (size_t)n * S_BOX * C_CLS);
#define PKJ(a, b) (int)(((unsigned)__builtin_amdgcn_readlane(j, (a)) & 0xFFFFu) | \
                        ((unsigned)__builtin_amdgcn_readlane(j, (b)) << 16))
        i32x4 g2 = { PKJ(0, 1),  PKJ(2, 3),   PKJ(4, 5),   PKJ(6, 7)  };
        i32x4 g3 = { PKJ(8, 9),  PKJ(10, 11), PKJ(12, 13), PKJ(14, 15) };
#undef PKJ
        tdm_issue(lds, ga, 1, C_CLS, S_BOX, C_CLS, 16, C_CLS, g2, g3);
    }
    __builtin_amdgcn_s_wait_tensorcnt(0);
    asm volatile("" ::: "memory");
    __syncthreads();

    const float* trow = &stage_t[wave][L * C_CLS];
    const float* srow = &stage_s[wave][L * C_CLS];
    const float invTau = 0.5f;    // 1/TAU

    // pass 1: row maxes (lane pair L / L+16 covers even/odd classes)
    float mt = -3.4e38f, ms = -3.4e38f;
    for (int c = h; c < C_CLS; c += 2) {
        mt = fmaxf(mt, trow[c]);
        ms = fmaxf(ms, srow[c]);
    }
    mt = fmaxf(mt, __shfl_xor(mt, 16, 32)) * invTau;
    ms = fmaxf(ms, __shfl_xor(ms, 16, 32)) * invTau;

    // pass 2: exp-sums
    float st = 0.0f, ss = 0.0f;
    for (int c = h; c < C_CLS; c += 2) {
        st += __expf(trow[c] * invTau - mt);
        ss += __expf(srow[c] * invTau - ms);
    }
    st += __shfl_xor(st, 16, 32);
    ss += __shfl_xor(ss, 16, 32);

    const float inv_st = 1.0f / st;
    const float lst = __logf(st), lss = __logf(ss);
    const float EPSF = 1e-6f;
    const float LOGEPS = -13.815510558f;   // log(1e-6)

    // confidence weight: max p_t == 1/sum_t
    float wgt = (inv_st - 0.7f) * (1.0f / 0.3f);
    wgt = fminf(fmaxf(wgt, 0.0f), 1.0f);

    // WMMA chain: A = p̂_t, B = log p̂_t - log p̂_s built on the fly;
    // lane (h*16+L) supplies (row L, K = 2h,2h+1) per the 16x4 f32 layout.
    v8f acc = {0.f, 0.f, 0.f, 0.f, 0.f, 0.f, 0.f, 0.f};
    #pragma unroll
    for (int k = 0; k < 23; ++k) {
        const int c = 4 * k + 2 * h;
        const bool v0 = (c < C_CLS), v1 = (c + 1 < C_CLS);
        float tl0 = v0 ? trow[c] : 0.0f;
        float tl1 = v1 ? trow[c + 1] : 0.0f;
        float sl0 = v0 ? srow[c] : 0.0f;
        float sl1 = v1 ? srow[c + 1] : 0.0f;
        float et0 = tl0 * invTau - mt, et1 = tl1 * invTau - mt;
        float es0 = sl0 * invTau - ms, es1 = sl1 * invTau - ms;
        float a0 = fmaxf(__expf(et0) * inv_st, EPSF);
        float a1 = fmaxf(__expf(et1) * inv_st, EPSF);
        float b0 = fmaxf(et0 - lst, LOGEPS) - fmaxf(es0 - lss, LOGEPS);
        float b1 = fmaxf(et1 - lst, LOGEPS) - fmaxf(es1 - lss, LOGEPS);
        v2f A; A.x = v0 ? a0 : 0.0f; A.y = v1 ? a1 : 0.0f;
        v2f B; B.x = v0 ? b0 : 0.0f; B.y = v1 ? b1 : 0.0f;
        acc = __builtin_amdgcn_wmma_f32_16x16x4_f32(
                  false, A, false, B, (short)0, acc, false, false);
    }

    // diag(D): (r,r) lives at VGPR (r&7), lane (r<8 ? r : r+16); spill via LDS
    // (intra-wave DS ops are in-order; no block barrier required).
    #pragma unroll
    for (int v = 0; v < 8; ++v) lds_d[wave][lane][v] = acc[v];
    __builtin_amdgcn_wave_barrier();
    const int src_lane = (L < 8) ? L : (L + 16);
    const float kl = lds_d[wave][src_lane][L & 7];

    const bool keep = (t_valid[n * T_BOX + t] != 0) &&
                      (best_iou[n * T_BOX + t] >= 0.5f);
    const float per = wgt * 4.0f * kl;   // TAU^2
    if (owner) {
        per_arr[n * T_BOX + t_raw]  = keep ? per : 0.0f;
        keep_arr[n * T_BOX + t_raw] = keep ? 1.0f : 0.0f;
    }
}

// ---------------- Phase C: per-image reduce ----------------
__global__ __launch_bounds__(256) void img_reduce_kernel(
    const float* __restrict__ per_arr, const float* __restrict__ keep_arr,
    float* __restrict__ loss_i, float* __restrict__ contrib)
{
    __shared__ float sh_s[256], sh_c[256];
    const int n = blockIdx.x;
    float s = 0.0f, c = 0.0f;
    for (int t = threadIdx.x; t < T_BOX; t += 256) {
        s += per_arr[n * T_BOX + t];
        c += keep_arr[n * T_BOX + t];
    }
    sh_s[threadIdx.x] = s; sh_c[threadIdx.x] = c;
    __syncthreads();
    for (int off = 128; off >= 1; off >>= 1) {
        if (threadIdx.x < off) {
            sh_s[threadIdx.x] += sh_s[threadIdx.x + off];
            sh_c[threadIdx.x] += sh_c[threadIdx.x + off];
        }
        __syncthreads();
    }
    if (threadIdx.x == 0) {
        float cnt = sh_c[0];
        float li  = sh_s[0] / fmaxf(cnt, 1.0f);
        loss_i[n]  = (cnt > 0.0f) ? li : 0.0f;
        contrib[n] = (cnt > 0.0f) ? 1.0f : 0.0f;
    }
}

// ---------------- Phase D: final scalar ----------------
__global__ __launch_bounds__(32) void final_kernel(
    const float* __restrict__ loss_i, const float* __restrict__ contrib,
    float* __restrict__ out)
{
    const int lane = threadIdx.x;
    float s = 0.0f, d = 0.0f;
    for (int i = lane; i < N_IMG; i += 32) { s += loss_i[i]; d += contrib[i]; }
    for (int off = 16; off >= 1; off >>= 1) {
        s += __shfl_xor(s, off, 32);
        d += __shfl_xor(d, off, 32);
    }
    if (lane == 0) out[0] = (d > 0.0f) ? (s / d) : 0.0f;
}

// ---------------- Launch ----------------
extern "C" void kernel_launch(void* const* d_in, const int* in_sizes, int n_in,
                              void* d_out, int out_size, void* d_ws, size_t ws_size,
                              hipStream_t stream) {
    const float* t_boxes  = (const float*)d_in[0];
    const float* t_logits = (const float*)d_in[1];
    const int*   t_valid  = (const int*)  d_in[2];
    const float* s_boxes  = (const float*)d_in[3];
    const float* s_logits = (const float*)d_in[4];
    const int*   s_valid  = (const int*)  d_in[5];
    float* out = (float*)d_out;

    const size_t NT = (size_t)N_IMG * T_BOX;
    char* ws = (char*)d_ws;
    float* best_iou = (float*)(ws);
    int*   best_j   = (int*)  (ws + NT * 4);
    float* per_arr  = (float*)(ws + NT * 8);
    float* keep_arr = (float*)(ws + NT * 12);
    float* loss_i   = (float*)(ws + NT * 16);
    float* contrib  = loss_i + N_IMG;

    iou_argmax_kernel<<<N_IMG * TCH, 256, 0, stream>>>(
        t_boxes, s_boxes, s_valid, best_iou, best_j);
    kd_wmma_kernel<<<N_IMG * KD_BLKS, 128, 0, stream>>>(
        t_logits, t_valid, s_logits, best_iou, best_j, per_arr, keep_arr);
    img_reduce_kernel<<<N_IMG, 256, 0, stream>>>(
        per_arr, keep_arr, loss_i, contrib);
    final_kernel<<<1, 32, 0, stream>>>(loss_i, contrib, out);
}